// DSFourierAttention_33208687132776
// MI455X (gfx1250) — compile-verified
//
#include <hip/hip_runtime.h>
#include <math.h>

// Problem constants (from reference): B=16, L=S=1024, H=8, E=64
#define BB 16
#define LL 1024
#define HH 8
#define EE 64
#define NF 513           // L/2 + 1 rfft bins
#define YP 528           // K/V freq padded to 33 * 16
#define YTILES 33
#define XPQ 640          // Q/out freq padded to 40 * 16 (8 waves x 5 blocks)
#define NBH (BB * HH)    // 128
#define BPBH 5           // blocks per (b,h); 8 strips each -> 40 strips

typedef __attribute__((ext_vector_type(2))) float v2f;
typedef __attribute__((ext_vector_type(8))) float v8f;

__device__ __forceinline__ v8f wmma_f32(v2f a, v2f b, v8f c) {
  // D = A(16x4 f32) * B(4x16 f32) + C(16x16 f32)
  return __builtin_amdgcn_wmma_f32_16x16x4_f32(false, a, false, b, (short)0, c,
                                               false, false);
}

// ---- CDNA5 async global->LDS copies (ASYNCcnt-tracked, no VGPR staging) ----
__device__ __forceinline__ void async_load_b128(void* lds, const void* gbase,
                                                int voff) {
  unsigned l = (unsigned)(uintptr_t)lds;  // flat LDS addr: low 32 bits = offset
  asm volatile("global_load_async_to_lds_b128 %0, %1, %2 offset:0"
               :
               : "v"(l), "v"(voff), "s"(gbase)
               : "memory");
}
__device__ __forceinline__ void async_load_b32(void* lds, const void* gbase,
                                               int voff) {
  unsigned l = (unsigned)(uintptr_t)lds;
  asm volatile("global_load_async_to_lds_b32 %0, %1, %2 offset:0"
               :
               : "v"(l), "v"(voff), "s"(gbase)
               : "memory");
}
__device__ __forceinline__ void wait_async0() {
  asm volatile("s_wait_asynccnt 0x0" ::: "memory");
}

// ---- 16-lane xor reductions via ds_swizzle (immediate pattern) ----
template <int MASK>
__device__ __forceinline__ float swz_xor(float x) {
  union { float f; int i; } u;
  u.f = x;
  u.i = __builtin_amdgcn_ds_swizzle(u.i, 0x1f | (MASK << 10));
  return u.f;
}
__device__ __forceinline__ float rowmax16(float v) {
  v = fmaxf(v, swz_xor<1>(v));
  v = fmaxf(v, swz_xor<2>(v));
  v = fmaxf(v, swz_xor<4>(v));
  v = fmaxf(v, swz_xor<8>(v));
  return v;
}
__device__ __forceinline__ float rowsum16(float v) {
  v += swz_xor<1>(v);
  v += swz_xor<2>(v);
  v += swz_xor<4>(v);
  v += swz_xor<8>(v);
  return v;
}

// ---------------------------------------------------------------------------
// Stockham radix-2 complex FFT, 1024 points, 256 threads, LDS double buffer.
// ---------------------------------------------------------------------------
__device__ __forceinline__ int fft1024(float* Sr0, float* Si0, float* Sr1,
                                       float* Si1, float sign) {
  float* Sr[2] = {Sr0, Sr1};
  float* Si[2] = {Si0, Si1};
  int cur = 0;
  int n = 1024;
  int st = 0;  // s = 1 << st
  while (n > 1) {
    const int m = n >> 1;
    const int s = 1 << st;
    for (int f = (int)threadIdx.x; f < 512; f += 256) {
      const int p = f >> st;
      const int qq = f & (s - 1);
      const float ang = sign * (6.28318530717958647692f * (float)p / (float)n);
      float sn, cs;
      __sincosf(ang, &sn, &cs);
      const int i0 = qq + s * p;
      const int i1 = i0 + s * m;
      const float ar = Sr[cur][i0], ai = Si[cur][i0];
      const float br = Sr[cur][i1], bi = Si[cur][i1];
      const float dr = ar - br, di = ai - bi;
      const int o0 = qq + ((s * p) << 1);
      const int o1 = o0 + s;
      Sr[cur ^ 1][o0] = ar + br;
      Si[cur ^ 1][o0] = ai + bi;
      Sr[cur ^ 1][o1] = dr * cs - di * sn;
      Si[cur ^ 1][o1] = dr * sn + di * cs;
    }
    __syncthreads();
    cur ^= 1;
    n = m;
    ++st;
  }
  return cur;
}

// ---------------------------------------------------------------------------
// Stage 1: rfft(q,k,v), ortho norm -> padded spectra in GEMM layouts.
//   QF[bh][x(640)][{re,im}][e]   (A-matrix for qk GEMM:     M=x, K=e)
//   KF[bh][{re,im}][e][y(528)]   (B-matrix for qk GEMM:     K=e, N=y)
//   VF[bh][y(528)][{re,im}][e]   (B-matrix for attn*v GEMM: K=y, N=e)
// ---------------------------------------------------------------------------
__global__ __launch_bounds__(256) void fourier_fwd_fft(
    const float* __restrict__ q, const float* __restrict__ k,
    const float* __restrict__ v, float* __restrict__ QF,
    float* __restrict__ KF, float* __restrict__ VF) {
  __shared__ float S0r[1024], S0i[1024], S1r[1024], S1i[1024];
  const int sid = blockIdx.x;
  const int which = sid / (NBH * EE);
  const int r = sid % (NBH * EE);
  const int bh = r / EE;
  const int e = r % EE;
  const int b = bh / HH;
  const int h = bh % HH;

  const float* src = (which == 0) ? q : ((which == 1) ? k : v);
  const float* sp = src + (size_t)b * LL * HH * EE + (size_t)h * EE + (size_t)e;
  // async gather of the strided time series directly into LDS
  #pragma unroll
  for (int i = 0; i < 4; ++i) {
    const int n = (int)threadIdx.x + i * 256;
    async_load_b32(&S0r[n], sp, n * HH * EE * 4);
    S0i[n] = 0.0f;
  }
  wait_async0();
  __syncthreads();
  const int cur = fft1024(S0r, S0i, S1r, S1i, -1.0f);
  float* Rr = cur ? S1r : S0r;
  float* Ri = cur ? S1i : S0i;
  const float scale = 0.03125f;  // 1/sqrt(1024), ortho

  if (which == 0) {
    float* dst = QF + (size_t)bh * XPQ * 2 * EE;
    for (int x = (int)threadIdx.x; x < XPQ; x += 256) {
      const float re = (x < NF) ? Rr[x] * scale : 0.0f;
      const float im = (x < NF) ? Ri[x] * scale : 0.0f;
      dst[((size_t)x * 2 + 0) * EE + e] = re;
      dst[((size_t)x * 2 + 1) * EE + e] = im;
    }
  } else if (which == 1) {
    float* dst = KF + (size_t)bh * 2 * EE * YP;
    for (int y = (int)threadIdx.x; y < YP; y += 256) {
      const float re = (y < NF) ? Rr[y] * scale : 0.0f;
      const float im = (y < NF) ? Ri[y] * scale : 0.0f;
      dst[(size_t)e * YP + y] = re;
      dst[(size_t)(EE + e) * YP + y] = im;
    }
  } else {
    float* dst = VF + (size_t)bh * YP * 2 * EE;
    for (int y = (int)threadIdx.x; y < YP; y += 256) {
      const float re = (y < NF) ? Rr[y] * scale : 0.0f;
      const float im = (y < NF) ? Ri[y] * scale : 0.0f;
      dst[((size_t)y * 2 + 0) * EE + e] = re;
      dst[((size_t)y * 2 + 1) * EE + e] = im;
    }
  }
}

// ---------------------------------------------------------------------------
// Stage 2: fused frequency-domain attention. 8 waves/block, one 16-row x
// strip per wave; K/V y-tiles staged to LDS with async B128 loads, double
// buffered; WMMA f32 for both GEMMs. OF[bh][{re,im}][e][x(640)].
// ---------------------------------------------------------------------------
__global__ __launch_bounds__(256) void fourier_attn_wmma(
    const float* __restrict__ QF, const float* __restrict__ KF,
    const float* __restrict__ VF, float* __restrict__ OF) {
  __shared__ float ktile[2][128 * 16];  // [buf][plane*64+e][y'] 8KB each
  __shared__ float vtile[2][16 * 128];  // [buf][y'][plane*64+e] 8KB each
  __shared__ float pbuf[8][16 * 17];
  const int flat = (int)threadIdx.x;
  const int lane = flat & 31;
  const int wave = flat >> 5;
  const int bh = (int)blockIdx.x / BPBH;
  const int strip = ((int)blockIdx.x % BPBH) * 8 + wave;  // 0..39, all valid
  const int x0 = strip * 16;
  const int lr = lane & 15;             // row (A) or col (B/C) within 16
  const int khalf = (lane >> 4) << 1;   // K pair base: 0 or 2
  const int rowoff = (lane >> 4) << 3;  // C-layout: rows j (+8 upper half)

  const float* kp = KF + (size_t)bh * 2 * EE * YP;
  const float* vp = VF + (size_t)bh * YP * 2 * EE;

  // A fragments: qf rows x0..x0+15 over full E (16 K-steps), re & im.
  v2f a_re[16], a_im[16];
  {
    const float* qp = QF + ((size_t)bh * XPQ + (size_t)(x0 + lr)) * 2 * EE;
    #pragma unroll
    for (int ks = 0; ks < 16; ++ks) {
      const int e0 = ks * 4 + khalf;
      a_re[ks] = v2f{qp[e0], qp[e0 + 1]};
      a_im[ks] = v2f{qp[EE + e0], qp[EE + e0 + 1]};
    }
  }

  v8f o_re[4], o_im[4];
  #pragma unroll
  for (int t = 0; t < 4; ++t) {
    o_re[t] = v8f{0.f, 0.f, 0.f, 0.f, 0.f, 0.f, 0.f, 0.f};
    o_im[t] = v8f{0.f, 0.f, 0.f, 0.f, 0.f, 0.f, 0.f, 0.f};
  }
  float mx[8], sm[8];
  #pragma unroll
  for (int j = 0; j < 8; ++j) {
    mx[j] = -3.4e38f;
    sm[j] = 0.0f;
  }

  // stage K/V tile for y0 into buffer `bsel` (whole workgroup cooperates)
  auto stage_kv = [&](int y0, int bsel) {
    #pragma unroll
    for (int i = 0; i < 2; ++i) {
      const int idx = i * 256 + flat;  // 0..511 16-byte chunks
      const int row = idx >> 2;        // 0..127 = plane*64+e
      const int seg = idx & 3;
      async_load_b128(&ktile[bsel][row * 16 + seg * 4], kp,
                      (row * YP + y0 + seg * 4) * 4);
      // V tile is 8KB contiguous starting at row y0
      async_load_b128(&vtile[bsel][idx * 4], vp, (y0 * 2 * EE + idx * 4) * 4);
    }
  };

  stage_kv(0, 0);
  wait_async0();
  __syncthreads();

  float* Lp = pbuf[wave];
  for (int yt = 0; yt < YTILES; ++yt) {
    const int y0 = yt * 16;
    const int buf = yt & 1;
    if (yt + 1 < YTILES) stage_kv(y0 + 16, buf ^ 1);
    const float* kb = ktile[buf];
    const float* vt = vtile[buf];

    // ---- qk tile: complex Gram via real WMMA chains over K=E ----
    v8f acc_re = v8f{0.f, 0.f, 0.f, 0.f, 0.f, 0.f, 0.f, 0.f};
    v8f acc_im = v8f{0.f, 0.f, 0.f, 0.f, 0.f, 0.f, 0.f, 0.f};
    #pragma unroll
    for (int ks = 0; ks < 16; ++ks) {
      const int e0 = ks * 4 + khalf;
      const v2f bkr = v2f{kb[e0 * 16 + lr], kb[(e0 + 1) * 16 + lr]};
      const v2f bki =
          v2f{kb[(EE + e0) * 16 + lr], kb[(EE + e0 + 1) * 16 + lr]};
      const v2f bkn = v2f{-bki.x, -bki.y};
      acc_re = wmma_f32(a_re[ks], bkr, acc_re);
      acc_re = wmma_f32(a_im[ks], bki, acc_re);
      acc_im = wmma_f32(a_im[ks], bkr, acc_im);
      acc_im = wmma_f32(a_re[ks], bkn, acc_im);
    }

    // ---- |qk|/sqrt(E) + online softmax (16-lane-half row reductions) ----
    const bool colValid = (y0 + lr) < NF;
    float p[8];
    #pragma unroll
    for (int j = 0; j < 8; ++j) {
      const float re = acc_re[j], im = acc_im[j];
      p[j] = colValid ? sqrtf(re * re + im * im) * 0.125f : -3.4e38f;
    }
    #pragma unroll
    for (int j = 0; j < 8; ++j) {
      const float vmax = rowmax16(p[j]);
      const float nm = fmaxf(mx[j], vmax);
      const float corr = __expf(mx[j] - nm);
      mx[j] = nm;
      const float pe = colValid ? __expf(p[j] - nm) : 0.0f;
      p[j] = pe;
      sm[j] = sm[j] * corr + rowsum16(pe);
      #pragma unroll
      for (int t = 0; t < 4; ++t) {
        o_re[t][j] *= corr;
        o_im[t][j] *= corr;
      }
    }

    // ---- transpose attn tile C-layout -> A-fragment via LDS (wave-private)
    #pragma unroll
    for (int j = 0; j < 8; ++j) Lp[(j + rowoff) * 17 + lr] = p[j];
    v2f ap[4];
    #pragma unroll
    for (int ks = 0; ks < 4; ++ks) {
      const int kk = ks * 4 + khalf;
      ap[ks] = v2f{Lp[lr * 17 + kk], Lp[lr * 17 + kk + 1]};
    }

    // ---- out[x,e] += attn * vf (re & im), 4 K-steps x 4 e-tiles ----
    #pragma unroll
    for (int ks = 0; ks < 4; ++ks) {
      const int yb = ks * 4 + khalf;  // local y row in the tile
      const float* v0 = vt + yb * 2 * EE;
      const float* v1 = vt + (yb + 1) * 2 * EE;
      #pragma unroll
      for (int t = 0; t < 4; ++t) {
        const int ee = t * 16 + lr;
        const v2f bvr = v2f{v0[ee], v1[ee]};
        const v2f bvi = v2f{v0[EE + ee], v1[EE + ee]};
        o_re[t] = wmma_f32(ap[ks], bvr, o_re[t]);
        o_im[t] = wmma_f32(ap[ks], bvi, o_im[t]);
      }
    }

    wait_async0();     // our staged loads for the next tile landed
    __syncthreads();   // everyone done reading `buf` & staging `buf^1`
  }

  // ---- normalize by softmax denominator and store spectrum ----
  float* op = OF + (size_t)bh * 2 * EE * XPQ;
  #pragma unroll
  for (int j = 0; j < 8; ++j) {
    const float inv = 1.0f / sm[j];
    const int x = x0 + j + rowoff;
    #pragma unroll
    for (int t = 0; t < 4; ++t) {
      const int e = t * 16 + lr;
      op[(size_t)e * XPQ + x] = o_re[t][j] * inv;
      op[(size_t)(EE + e) * XPQ + x] = o_im[t][j] * inv;
    }
  }
}

// ---------------------------------------------------------------------------
// Stage 3: hermitian-extend, inverse FFT (ortho), fuse *tau + delta and the
// [B,H,E,L] -> [B,L,H,E] transpose on store.
// ---------------------------------------------------------------------------
__global__ __launch_bounds__(256) void fourier_inv_fft(
    const float* __restrict__ OF, const float* __restrict__ tau,
    const float* __restrict__ delta, float* __restrict__ out) {
  __shared__ float S0r[1024], S0i[1024], S1r[1024], S1i[1024];
  const int sid = blockIdx.x;
  const int bh = sid / EE;
  const int e = sid % EE;
  const int b = bh / HH;
  const int h = bh % HH;

  const float* re = OF + (size_t)bh * 2 * EE * XPQ + (size_t)e * XPQ;
  const float* im = re + (size_t)EE * XPQ;
  for (int kk = (int)threadIdx.x; kk < 1024; kk += 256) {
    const int k2 = (kk <= 512) ? kk : (1024 - kk);
    const float sg = (kk <= 512) ? 1.0f : -1.0f;
    S0r[kk] = re[k2];
    S0i[kk] = sg * im[k2];
  }
  __syncthreads();
  const int cur = fft1024(S0r, S0i, S1r, S1i, +1.0f);
  float* Rr = cur ? S1r : S0r;
  const float scale = 0.03125f;  // 1/sqrt(1024), ortho
  const float t = tau[b];
  for (int l = (int)threadIdx.x; l < LL; l += 256) {
    const float val = Rr[l] * scale;
    out[(((size_t)b * LL + l) * HH + h) * EE + e] =
        val * t + delta[(size_t)b * LL + l];
  }
}

// ---------------------------------------------------------------------------
extern "C" void kernel_launch(void* const* d_in, const int* in_sizes, int n_in,
                              void* d_out, int out_size, void* d_ws,
                              size_t ws_size, hipStream_t stream) {
  (void)in_sizes; (void)n_in; (void)out_size; (void)ws_size;
  const float* q = (const float*)d_in[0];
  const float* k = (const float*)d_in[1];
  const float* v = (const float*)d_in[2];
  // d_in[3] = mask (unused by the reference module)
  const float* tau = (const float*)d_in[4];
  const float* delta = (const float*)d_in[5];
  float* out = (float*)d_out;

  float* ws = (float*)d_ws;
  const size_t arrQ = (size_t)NBH * XPQ * 2 * EE;  // QF / OF
  const size_t arrK = (size_t)NBH * YP * 2 * EE;   // KF / VF
  float* QF = ws;
  float* KF = QF + arrQ;
  float* VF = KF + arrK;
  float* OF = VF + arrK;

  hipLaunchKernelGGL(fourier_fwd_fft, dim3(3 * NBH * EE), dim3(256), 0, stream,
                     q, k, v, QF, KF, VF);
  hipLaunchKernelGGL(fourier_attn_wmma, dim3(NBH * BPBH), dim3(256), 0, stream,
                     QF, KF, VF, OF);
  hipLaunchKernelGGL(fourier_inv_fft, dim3(NBH * EE), dim3(256), 0, stream, OF,
                     tau, delta, out);
}